// GraphIsomorphismNetwork_77979426226450
// MI455X (gfx1250) — compile-verified
//
#include <hip/hip_runtime.h>
#include <hip/hip_bf16.h>
#include <math.h>

#define NV 4096
#define NE 8192
#define ADIM 64
#define BDIM 32
#define HDIM 128
#define NLAYER 4

typedef float v2f __attribute__((ext_vector_type(2)));
typedef float v8f __attribute__((ext_vector_type(8)));

// ---------------------------------------------------------------------------
// Recover src[e]/dst[e] from the one-hot incidence matrix [NV, NE] row-major.
// Every column has exactly one nonzero, so every idx_out slot gets written.
// Bandwidth-bound scan: float4 loads, conditional scatter of the row index.
// ---------------------------------------------------------------------------
__global__ __launch_bounds__(256) void extract_index_kernel(
    const float* __restrict__ mat, int* __restrict__ idx_out) {
  long t = (long)blockIdx.x * blockDim.x + threadIdx.x;  // group of 4 elements
  float4 val = ((const float4*)mat)[t];
  long base = t << 2;
  int v = (int)(base >> 13);    // / NE (8192)
  int e = (int)(base & (NE - 1));
  if (val.x != 0.f) idx_out[e]     = v;
  if (val.y != 0.f) idx_out[e + 1] = v;
  if (val.z != 0.f) idx_out[e + 2] = v;
  if (val.w != 0.f) idx_out[e + 3] = v;
}

// ---------------------------------------------------------------------------
// C[m,n] = act( sum_k A[m,k] * W[n,k] + bias[n] )
// A: [M,K] row-major, W: [N,K] row-major (torch Linear weight layout).
// One 16x16 output tile per wave32; K-loop of V_WMMA_F32_16X16X4_F32.
// ACT: 0 = leaky_relu(0.01), 1 = tanh.
// Per ISA 7.12.2 (lane = 16*half + lm):
//   A frag elem i  = A[lm, k0 + 2*half + i]        -> float2 load
//   B frag elem i  = B[k0 + 2*half + i, lm] = W[lm, k0+2*half+i] -> float2 load
//   C/D elem i     = C[8*half + i, lm]
// ---------------------------------------------------------------------------
template <int ACT>
__global__ __launch_bounds__(256) void wmma_gemm_bias_act(
    const float* __restrict__ A, const float* __restrict__ W,
    const float* __restrict__ bias, float* __restrict__ C,
    int M, int N, int K) {
  const int wave = threadIdx.x >> 5;
  const int lane = threadIdx.x & 31;
  const int half = lane >> 4;
  const int lm = lane & 15;
  const int ntn = N >> 4;
  const int tile = blockIdx.x * 8 + wave;
  const int tm = tile / ntn;
  const int tn = tile % ntn;

  const float* Arow = A + (long)(tm * 16 + lm) * K + 2 * half;
  const float* Wrow = W + (long)(tn * 16 + lm) * K + 2 * half;

  v8f acc = {0.f, 0.f, 0.f, 0.f, 0.f, 0.f, 0.f, 0.f};
  for (int k0 = 0; k0 < K; k0 += 4) {
    v2f a = *(const v2f*)(Arow + k0);
    v2f b = *(const v2f*)(Wrow + k0);
    // D = A(16x4) x B(4x16) + C  (fp32 WMMA, wave32)
    acc = __builtin_amdgcn_wmma_f32_16x16x4_f32(
        /*neg_a=*/false, a, /*neg_b=*/false, b,
        /*c_mod=*/(short)0, acc, /*reuse_a=*/false, /*reuse_b=*/false);
  }

  const float bn = bias[tn * 16 + lm];
  float* Cp = C + (long)(tm * 16 + 8 * half) * N + tn * 16 + lm;
#pragma unroll
  for (int i = 0; i < 8; ++i) {
    float x = acc[i] + bn;
    if (ACT == 0) {
      x = (x > 0.f) ? x : 0.01f * x;
    } else {
      x = tanhf(x);
    }
    Cp[(long)i * N] = x;
  }
}

// ---------------------------------------------------------------------------
// Per edge e: t = relu(he[e,:] + hv[dst[e],:]); hv_acc[src[e],:] += t
// (hv_acc pre-initialized to hv, so this realizes hv + vew1 @ relu(he + vew2^T hv))
// ---------------------------------------------------------------------------
__global__ __launch_bounds__(HDIM) void edge_relu_scatter(
    const float* __restrict__ he, const float* __restrict__ hv,
    float* __restrict__ hv_acc, const int* __restrict__ src,
    const int* __restrict__ dst) {
  const int e = blockIdx.x;
  const int j = threadIdx.x;
  const int s = src[e];
  const int d = dst[e];
  float t = he[(long)e * HDIM + j] + hv[(long)d * HDIM + j];
  t = fmaxf(t, 0.f);
  atomicAdd(hv_acc + (long)s * HDIM + j, t);  // global_atomic_add_f32
}

// ---------------------------------------------------------------------------
// he_out[e] = concat(hv[src[e]], hv[dst[e]], he[e])   -> [NE, 3H]
// ---------------------------------------------------------------------------
__global__ __launch_bounds__(HDIM) void he_out_gather(
    const float* __restrict__ hv, const float* __restrict__ he,
    const int* __restrict__ src, const int* __restrict__ dst,
    float* __restrict__ out) {
  const int e = blockIdx.x;
  const int j = threadIdx.x;
  float* o = out + (long)e * (3 * HDIM);
  o[j] = hv[(long)src[e] * HDIM + j];
  o[HDIM + j] = hv[(long)dst[e] * HDIM + j];
  o[2 * HDIM + j] = he[(long)e * HDIM + j];
}

extern "C" void kernel_launch(void* const* d_in, const int* in_sizes, int n_in,
                              void* d_out, int out_size, void* d_ws,
                              size_t ws_size, hipStream_t stream) {
  const float* atom_ftr = (const float*)d_in[0];
  const float* bond_ftr = (const float*)d_in[1];
  const float* vew1 = (const float*)d_in[2];
  const float* vew2 = (const float*)d_in[3];
  const float* vW1 = (const float*)d_in[4];
  const float* vb1 = (const float*)d_in[5];
  const float* vW2 = (const float*)d_in[6];
  const float* vb2 = (const float*)d_in[7];
  const float* eW1 = (const float*)d_in[8];
  const float* eb1 = (const float*)d_in[9];
  const float* eW2 = (const float*)d_in[10];
  const float* eb2 = (const float*)d_in[11];
  const float* mlp_W = (const float*)d_in[12];  // [NLAYER, H, H]
  const float* mlp_b = (const float*)d_in[13];  // [NLAYER, H]

  // Workspace layout
  char* ws = (char*)d_ws;
  int* src = (int*)ws;              // NE ints
  int* dst = src + NE;              // NE ints
  float* t0 = (float*)(ws + 2 * NE * sizeof(int));  // up to NE*H floats
  float* hvA = t0 + (size_t)NE * HDIM;              // NV*H
  float* hvB = hvA + (size_t)NV * HDIM;             // NV*H
  float* he = hvB + (size_t)NV * HDIM;              // NE*H

  // 1) Index recovery from one-hot incidence matrices (bandwidth-dominant).
  {
    const int nblk = (NV * (NE / 4)) / 256;  // 32768
    extract_index_kernel<<<nblk, 256, 0, stream>>>(vew1, src);
    extract_index_kernel<<<nblk, 256, 0, stream>>>(vew2, dst);
  }

  const int gv = (NV / 16) * (HDIM / 16) / 8;  // 256 blocks, 8 waves each
  const int ge = (NE / 16) * (HDIM / 16) / 8;  // 512 blocks

  // 2) hv = tanh(leaky(atom @ vW1^T + vb1) @ vW2^T + vb2)
  wmma_gemm_bias_act<0><<<gv, 256, 0, stream>>>(atom_ftr, vW1, vb1, t0, NV, HDIM, ADIM);
  wmma_gemm_bias_act<1><<<gv, 256, 0, stream>>>(t0, vW2, vb2, hvA, NV, HDIM, HDIM);

  // 3) he = tanh(leaky(bond @ eW1^T + eb1) @ eW2^T + eb2)
  wmma_gemm_bias_act<0><<<ge, 256, 0, stream>>>(bond_ftr, eW1, eb1, t0, NE, HDIM, BDIM);
  wmma_gemm_bias_act<1><<<ge, 256, 0, stream>>>(t0, eW2, eb2, he, NE, HDIM, HDIM);

  // 4) Message-passing layers
  float* hv_out = (float*)d_out;  // final hv lands directly in d_out[0 : NV*H)
  for (int l = 0; l < NLAYER; ++l) {
    hipMemcpyAsync(hvB, hvA, (size_t)NV * HDIM * sizeof(float),
                   hipMemcpyDeviceToDevice, stream);
    edge_relu_scatter<<<NE, HDIM, 0, stream>>>(he, hvA, hvB, src, dst);
    float* out = (l == NLAYER - 1) ? hv_out : hvA;
    wmma_gemm_bias_act<0><<<gv, 256, 0, stream>>>(
        hvB, mlp_W + (size_t)l * HDIM * HDIM, mlp_b + (size_t)l * HDIM, out,
        NV, HDIM, HDIM);
  }

  // 5) he_out = [hv[src] | hv[dst] | he]  -> d_out[NV*H : NV*H + NE*3H)
  he_out_gather<<<NE, HDIM, 0, stream>>>(hv_out, he, src, dst,
                                         hv_out + (size_t)NV * HDIM);
}